// GAT_18296560681320
// MI455X (gfx1250) — compile-verified
//
#include <hip/hip_runtime.h>
#include <hip/hip_bf16.h>

#define D_FEAT 64
#define D_OUT  128

typedef __attribute__((ext_vector_type(2))) float v2f;
typedef __attribute__((ext_vector_type(8))) float v8f;

// Order-preserving f32 <-> u32 encoding so unsigned atomicMax == float max.
__device__ __forceinline__ unsigned enc_f32(float f) {
    unsigned u = __float_as_uint(f);
    return (u & 0x80000000u) ? ~u : (u | 0x80000000u);
}
__device__ __forceinline__ float dec_f32(unsigned u) {
    unsigned v = (u & 0x80000000u) ? (u & 0x7fffffffu) : ~u;
    return __uint_as_float(v);
}

// ---------------------------------------------------------------- init
__global__ void gat_init(float* __restrict__ rst, float* __restrict__ segsum,
                         unsigned* __restrict__ segmax, int n_nodes) {
    size_t i      = (size_t)blockIdx.x * blockDim.x + threadIdx.x;
    size_t stride = (size_t)gridDim.x * blockDim.x;
    size_t total  = (size_t)n_nodes * D_FEAT;
    for (size_t j = i; j < total; j += stride) rst[j] = 0.0f;
    for (size_t j = i; j < (size_t)n_nodes; j += stride) {
        segsum[j] = 0.0f;
        segmax[j] = 0u;   // 0 < enc(-inf): identity for the encoded max
    }
}

// ------------------------------------------------- per-edge dot + seg max
// One wave (32 lanes) per edge; each lane handles 2 features via float2.
__global__ __launch_bounds__(256) void gat_score(
    const float* __restrict__ hk, const float* __restrict__ hu,
    const int* __restrict__ src, const int* __restrict__ dst,
    float* __restrict__ scores, unsigned* __restrict__ segmax, int n_edges) {
    int e = blockIdx.x * 8 + (threadIdx.x >> 5);   // wave-uniform
    if (e >= n_edges) return;
    int lane = threadIdx.x & 31;
    int s = src[e], d = dst[e];
    float2 a = ((const float2*)(hk + (size_t)s * D_FEAT))[lane];
    float2 b = ((const float2*)(hu + (size_t)d * D_FEAT))[lane];
    float p = a.x * b.x + a.y * b.y;
#pragma unroll
    for (int off = 16; off > 0; off >>= 1) p += __shfl_xor(p, off, 32);
    if (lane == 0) {
        scores[e] = p;
        atomicMax(segmax + d, enc_f32(p));
    }
}

// ------------------------------------------------- exp(score - max), seg sum
__global__ __launch_bounds__(256) void gat_exp(
    const int* __restrict__ dst, float* __restrict__ scores,
    const unsigned* __restrict__ segmax, float* __restrict__ segsum, int n_edges) {
    int e = blockIdx.x * blockDim.x + threadIdx.x;
    if (e >= n_edges) return;
    int d = dst[e];
    float ex = expf(scores[e] - dec_f32(segmax[d]));
    scores[e] = ex;
    unsafeAtomicAdd(segsum + d, ex);   // native global_atomic_add_f32
}

// ------------------------------------------------- alpha * hk[src] -> rst[dst]
__global__ __launch_bounds__(256) void gat_scatter(
    const float* __restrict__ hk,
    const int* __restrict__ src, const int* __restrict__ dst,
    const float* __restrict__ scores, const float* __restrict__ segsum,
    float* __restrict__ rst, int n_edges) {
    int e = blockIdx.x * 8 + (threadIdx.x >> 5);   // wave-uniform
    if (e >= n_edges) return;
    int lane = threadIdx.x & 31;
    int s = src[e], d = dst[e];
    float alpha = scores[e] / segsum[d];
    float2 a = ((const float2*)(hk + (size_t)s * D_FEAT))[lane];
    float* r = rst + (size_t)d * D_FEAT + lane * 2;
    unsafeAtomicAdd(r,     alpha * a.x);
    unsafeAtomicAdd(r + 1, alpha * a.y);
}

// ------------------------------------------------- out = relu(rst @ W^T + b)
// fp32 WMMA 16x16x4: block = 8 waves; block covers 16 rows, each wave one
// 16-wide column tile (8*16 = 128 = D_OUT). 16 WMMAs sweep K = 64.
__global__ __launch_bounds__(256) void gat_fc_relu_wmma(
    const float* __restrict__ rst, const float* __restrict__ W,
    const float* __restrict__ bias, float* __restrict__ out, int n_nodes) {
    int m0   = blockIdx.x * 16;
    int lane = threadIdx.x & 31;
    int half = lane >> 4;          // 0: lanes 0-15, 1: lanes 16-31
    int l15  = lane & 15;
    int n0   = (threadIdx.x >> 5) * 16;

    int arow_i = m0 + l15;
    if (arow_i >= n_nodes) arow_i = n_nodes - 1;   // clamp: keep EXEC all-ones
    const float* arow = rst + (size_t)arow_i * D_FEAT;
    const float* brow = W   + (size_t)(n0 + l15) * D_FEAT;   // B[k][n] = W[n][k]

    v8f acc = {};
#pragma unroll
    for (int k = 0; k < 16; ++k) {
        int kb = k * 4 + half * 2;       // A: lanes 0-15 -> K=4k,4k+1 ; 16-31 -> 4k+2,4k+3
        v2f a, b;
        a[0] = arow[kb]; a[1] = arow[kb + 1];
        b[0] = brow[kb]; b[1] = brow[kb + 1];
        acc = __builtin_amdgcn_wmma_f32_16x16x4_f32(
            /*neg_a=*/false, a, /*neg_b=*/false, b,
            /*c_mod=*/(short)0, acc, /*reuse_a=*/false, /*reuse_b=*/false);
    }

    float bv = bias[n0 + l15];
#pragma unroll
    for (int r = 0; r < 8; ++r) {
        int m = m0 + r + 8 * half;       // C/D layout: VGPR r -> M=r (lo half) / r+8 (hi half)
        if (m < n_nodes) {
            float v = acc[r] + bv;
            out[(size_t)m * D_OUT + n0 + l15] = v > 0.0f ? v : 0.0f;
        }
    }
}

extern "C" void kernel_launch(void* const* d_in, const int* in_sizes, int n_in,
                              void* d_out, int out_size, void* d_ws, size_t ws_size,
                              hipStream_t stream) {
    const float* hk = (const float*)d_in[0];
    const float* hu = (const float*)d_in[1];
    const float* W  = (const float*)d_in[2];
    const float* b  = (const float*)d_in[3];
    const int* src  = (const int*)d_in[4];
    const int* dst  = (const int*)d_in[5];
    int n_nodes = in_sizes[0] / D_FEAT;
    int n_edges = in_sizes[4];
    float* out = (float*)d_out;

    // Workspace carve-out (256B-aligned slices)
    char* ws = (char*)d_ws;
    size_t off = 0;
    auto carve = [&](size_t bytes) -> void* {
        void* p = ws + off;
        off = (off + bytes + 255) & ~(size_t)255;
        return p;
    };
    float*    scores = (float*)   carve((size_t)n_edges * sizeof(float));
    unsigned* segmax = (unsigned*)carve((size_t)n_nodes * sizeof(unsigned));
    float*    segsum = (float*)   carve((size_t)n_nodes * sizeof(float));
    float*    rst    = (float*)   carve((size_t)n_nodes * D_FEAT * sizeof(float));
    (void)ws_size; (void)n_in; (void)out_size;

    int edge_blocks = (n_edges + 7) / 8;   // 8 waves/block, wave per edge

    gat_init<<<1024, 256, 0, stream>>>(rst, segsum, segmax, n_nodes);
    gat_score<<<edge_blocks, 256, 0, stream>>>(hk, hu, src, dst, scores, segmax, n_edges);
    gat_exp<<<(n_edges + 255) / 256, 256, 0, stream>>>(dst, scores, segmax, segsum, n_edges);
    gat_scatter<<<edge_blocks, 256, 0, stream>>>(hk, src, dst, scores, segsum, rst, n_edges);
    gat_fc_relu_wmma<<<(n_nodes + 15) / 16, 256, 0, stream>>>(rst, W, b, out, n_nodes);
}